// Encoder_11974368821732
// MI455X (gfx1250) — compile-verified
//
#include <hip/hip_runtime.h>
#include <hip/hip_bf16.h>

// GraphSAGE encoder, algebraically fused and bias-folded:
//   AB[0:5][d]  = sum_c W1[f][c] * W[c][d]        (self path)
//   AB[5:10][d] = sum_c W1[f][c] * W[128+c][d]    (neighbor path)
//   AB[10][d]   = b[d]   (bias row, paired with constant-1.0 A column)
//   AB[11][d]   = 0      (pad)
//   y[s,n,d] = relu( [self(5)|neigh(5)|1|0] . AB[:,d] )
// One wave32 per 16-node tile; V_WMMA_F32_16X16X4_F32, K=12 -> 3 k-steps.

#define SS   8
#define NN   50000
#define FF   5
#define DEGG 32
#define DOUT 128
#define TILES 3125           // NN / 16, exact

typedef float v2f __attribute__((ext_vector_type(2)));
typedef float v8f __attribute__((ext_vector_type(8)));

// ---------- Kernel 1: fuse W1 (5x128) with W (256x128) + bias into AB (12x128) ----------
__global__ __launch_bounds__(128) void sage_precompute(const float* __restrict__ W1,
                                                       const float* __restrict__ W,
                                                       const float* __restrict__ b,
                                                       float* __restrict__ AB) {
    const int d = threadIdx.x;                 // output column 0..127
    float acc[10];
#pragma unroll
    for (int i = 0; i < 10; ++i) acc[i] = 0.0f;
    for (int c = 0; c < 128; ++c) {
        const float wt = W[c * DOUT + d];            // top half of W
        const float wb = W[(128 + c) * DOUT + d];    // bottom half of W
#pragma unroll
        for (int f = 0; f < FF; ++f) {
            const float w1 = W1[f * 128 + c];        // scalar broadcast
            acc[f]     += w1 * wt;
            acc[5 + f] += w1 * wb;
        }
    }
#pragma unroll
    for (int i = 0; i < 10; ++i) AB[i * DOUT + d] = acc[i];
    AB[10 * DOUT + d] = b[d];                  // bias row (A column K=10 is 1.0)
    AB[11 * DOUT + d] = 0.0f;                  // pad row
}

// ---------- Kernel 2: gather + mean + fused GEMM via WMMA f32 16x16x4 ----------
__global__ __launch_bounds__(256) void sage_wmma(const float* __restrict__ x,
                                                 const int*   __restrict__ nidx,
                                                 const float* __restrict__ AB,
                                                 float* __restrict__ out) {
    const int lane = threadIdx.x & 31;
    const int tile = blockIdx.x * 8 + (threadIdx.x >> 5);   // 0..24999, exact fit
    const int s    = tile / TILES;
    const int nt   = tile - s * TILES;
    const int node_base = nt * 16;

    const int half = lane >> 4;        // 0: lanes 0-15, 1: lanes 16-31
    const int m    = lane & 15;        // row within tile / col within N-tile
    const int node = node_base + m;

    const float* xs = x + (size_t)s * (NN * FF);

    // Self features (both half-waves load the same 5 floats; L2-hot).
    float sf[FF];
#pragma unroll
    for (int f = 0; f < FF; ++f) sf[f] = xs[node * FF + f];

    // Neighbor mean: lane pair (m, m+16) splits the 32 neighbors 16/16.
    const int* ids = nidx + ((size_t)s * NN + node) * DEGG + half * 16;
    float ns[FF] = {0.f, 0.f, 0.f, 0.f, 0.f};
#pragma unroll 4
    for (int k = 0; k < 16; ++k) {
        const float* xp = xs + (size_t)ids[k] * FF;
#pragma unroll
        for (int f = 0; f < FF; ++f) ns[f] += xp[f];
    }
#pragma unroll
    for (int f = 0; f < FF; ++f)
        ns[f] = (ns[f] + __shfl_xor(ns[f], 16, 32)) * (1.0f / DEGG);

    // A fragments (16x4 f32, K columns): half=0 holds K{0,1 | 4,5 | 8,9},
    // half=1 holds K{2,3 | 6,7 | 10,11}. K layout: 0-4=self, 5-9=neigh, 10=1.0, 11=0.
    // Explicit 2-way selects (6 v_cndmask) -- no dynamic private-array indexing.
    const bool h = (half != 0);
    v2f a0, a1, a2;
    a0.x = h ? sf[2] : sf[0];
    a0.y = h ? sf[3] : sf[1];
    a1.x = h ? ns[1] : sf[4];
    a1.y = h ? ns[2] : ns[0];
    a2.x = h ? 1.0f  : ns[3];   // K=10: bias multiplier
    a2.y = h ? 0.0f  : ns[4];   // K=11: pad

    const int kb = 2 * half;
#pragma unroll
    for (int j = 0; j < 8; ++j) {
        const int col = 16 * j + m;

        // B fragments (4x16 f32): row-striped across lanes, mirrors A halving.
        v2f b0, b1, b2;
        b0.x = AB[(0 + kb) * DOUT + col]; b0.y = AB[(1 + kb) * DOUT + col];
        b1.x = AB[(4 + kb) * DOUT + col]; b1.y = AB[(5 + kb) * DOUT + col];
        b2.x = AB[(8 + kb) * DOUT + col]; b2.y = AB[(9 + kb) * DOUT + col];

        v8f c = {};   // zero accumulator -> inline SRC2=0 on first WMMA
        c = __builtin_amdgcn_wmma_f32_16x16x4_f32(false, a0, false, b0, (short)0, c, false, false);
        c = __builtin_amdgcn_wmma_f32_16x16x4_f32(false, a1, false, b1, (short)0, c, false, false);
        c = __builtin_amdgcn_wmma_f32_16x16x4_f32(false, a2, false, b2, (short)0, c, false, false);

        // D layout: VGPR r -> row (r + 8*half), column = col. relu on store.
        float* op = out + ((size_t)s * NN + node_base + 8 * half) * DOUT + col;
#pragma unroll
        for (int r = 0; r < 8; ++r)
            op[(size_t)r * DOUT] = fmaxf(c[r], 0.0f);
    }
}

extern "C" void kernel_launch(void* const* d_in, const int* in_sizes, int n_in,
                              void* d_out, int out_size, void* d_ws, size_t ws_size,
                              hipStream_t stream) {
    const float* x    = (const float*)d_in[0];   // [S,N,F]
    const int*   nidx = (const int*)  d_in[1];   // [S,N,DEG]
    const float* W1   = (const float*)d_in[2];   // [F,128]
    const float* W    = (const float*)d_in[3];   // [256,128]
    const float* b    = (const float*)d_in[4];   // [128]
    float*       out  = (float*)d_out;           // [S,N,128]
    float*       AB   = (float*)d_ws;            // 12*128 floats = 6 KB scratch

    sage_precompute<<<1, 128, 0, stream>>>(W1, W, b, AB);
    sage_wmma<<<(SS * TILES) / 8, 256, 0, stream>>>(x, nidx, AB, out);
}